// att_rnn_7516192768288
// MI455X (gfx1250) — compile-verified
//
#include <hip/hip_runtime.h>
#include <math.h>

typedef __attribute__((ext_vector_type(2))) float v2f;
typedef __attribute__((ext_vector_type(4))) float v4f;
typedef __attribute__((ext_vector_type(8))) float v8f;

#define KC 16  // K-chunk staged in LDS per iteration (4 wmma k-steps)

// ---------------------------------------------------------------------------
// GEMM inner loop (fp32 WMMA 16x16x4), compile-time strides.
// Block = 128 threads (4 waves). A-tile (16 x KC) staged once per chunk and
// shared by all 4 waves; each wave stages its own 16x16 B-tile. LDS rows are
// padded to 20 floats (80 B): keeps 16-B alignment for b128 LDS stores and
// spreads bank access (gcd(20,64)=4 -> 16 distinct banks over 16 rows).
//
// Fragment layouts (V_WMMA_F32_16X16X4_F32):
//   A: lane l: m=l%16, half=l/16; a.x=A[m][2*half],  a.y=A[m][2*half+1]
//   B: lane l: n=l%16;            b.x=B[2*half][n],  b.y=B[2*half+1][n]
//   C/D: v8f, c[r] @ lane l = D[r + 8*(l/16)][l%16]
// ---------------------------------------------------------------------------
template <int LDA, int LDW>
__device__ inline void gemm_tile_acc(const float* __restrict__ A,
                                     const float* __restrict__ W,
                                     int K, int m0, int n0w,
                                     float (*ldsA)[20], float (*ldsB)[20],
                                     v8f& acc)
{
    const int t    = threadIdx.x;
    const int lane = t & 31;
    const int half = lane >> 4;
    const int mrow = lane & 15;

    // A staging: 256 floats / 128 threads -> one float2 per thread, coalesced.
    const int ar = t >> 3;          // row 0..15
    const int ac = (t & 7) * 2;     // col 0,2,..,14
    // B staging (per wave): 256 floats / 32 lanes -> two float4 per lane.
    const int br = lane >> 1;       // row 0..15
    const int bc = (lane & 1) * 8;  // col 0 or 8

    const float* aptr = A + (size_t)(m0 + ar) * LDA + ac;
    const float* bptr = W + (size_t)br * LDW + n0w + bc;

    for (int k0 = 0; k0 < K; k0 += KC) {
        const v2f a2 = *(const v2f*)(aptr + k0);
        ldsA[ar][ac]     = a2.x;
        ldsA[ar][ac + 1] = a2.y;

        const v4f b0 = *(const v4f*)(bptr + (size_t)k0 * LDW);
        const v4f b1 = *(const v4f*)(bptr + (size_t)k0 * LDW + 4);
        *(v4f*)&ldsB[br][bc]     = b0;
        *(v4f*)&ldsB[br][bc + 4] = b1;
        __syncthreads();

#pragma unroll
        for (int ks = 0; ks < KC; ks += 4) {
            v2f a, b;
            a.x = ldsA[mrow][ks + 2 * half];
            a.y = ldsA[mrow][ks + 2 * half + 1];
            b.x = ldsB[ks + 2 * half][mrow];
            b.y = ldsB[ks + 2 * half + 1][mrow];
            acc = __builtin_amdgcn_wmma_f32_16x16x4_f32(
                false, a, false, b, (short)0, acc, false, false);
        }
        __syncthreads();
    }
}

// ---------------------------------------------------------------------------
// Fused LSTM cell: z = A1@W + A2@Uw + bias  (W,Uw: [K,4U], gate order i,f,g,o)
// Wave w computes the 16x16 tile of gate w for the same (m,n) block; gates
// combined through LDS: c' = sig(f)*c + sig(i)*tanh(g); h' = sig(o)*tanh(c')
// Grid: (U/16, B/16), block 128.
// ---------------------------------------------------------------------------
__global__ void __launch_bounds__(128)
lstm_cell_kernel(const float* __restrict__ A1, const float* __restrict__ W,
                 const float* __restrict__ A2, const float* __restrict__ Uw,
                 const float* __restrict__ bias, const float* __restrict__ Cprev,
                 float* __restrict__ Hout, float* __restrict__ Cout,
                 int K1, int K2, int Uu)
{
    __shared__ float ldsA[16][20];
    __shared__ float ldsB[4][16][20];
    __shared__ float gbuf[4][16][17];

    const int t    = threadIdx.x;
    const int wave = t >> 5;        // gate id: 0=i 1=f 2=g 3=o
    const int lane = t & 31;
    const int half = lane >> 4;
    const int mrow = lane & 15;
    const int m0   = blockIdx.y * 16;
    const int n0   = blockIdx.x * 16;
    const int n0w  = wave * Uu + n0;   // column base into [K,4U] weights

    v8f acc = {};
    gemm_tile_acc<1024, 4096>(A1, W,  K1, m0, n0w, ldsA, ldsB[wave], acc);
    gemm_tile_acc<1024, 4096>(A2, Uw, K2, m0, n0w, ldsA, ldsB[wave], acc);

    const float bv = bias[n0w + mrow];
#pragma unroll
    for (int r = 0; r < 8; ++r) {
        const float z = acc[r] + bv;
        const float a = (wave == 2) ? tanhf(z) : (1.0f / (1.0f + __expf(-z)));
        gbuf[wave][r + 8 * half][mrow] = a;
    }
    __syncthreads();

#pragma unroll
    for (int e = t; e < 256; e += 128) {
        const int m = e >> 4, n = e & 15;
        const float iv = gbuf[0][m][n];
        const float fv = gbuf[1][m][n];
        const float gv = gbuf[2][m][n];
        const float ov = gbuf[3][m][n];
        const size_t idx = (size_t)(m0 + m) * Uu + n0 + n;
        const float cn = fv * Cprev[idx] + iv * gv;
        Cout[idx] = cn;
        Hout[idx] = ov * tanhf(cn);
    }
}

// ---------------------------------------------------------------------------
// s = h2 @ Ws + bs  (128x1024 @ 1024x1024). Block = 4 waves -> 16 x 64 tile,
// shared A staging. Grid: (U/64, B/16), block 128.
// ---------------------------------------------------------------------------
__global__ void __launch_bounds__(128)
sproj_kernel(const float* __restrict__ H2, const float* __restrict__ Ws,
             const float* __restrict__ bs, float* __restrict__ Sout, int Uu)
{
    __shared__ float ldsA[16][20];
    __shared__ float ldsB[4][16][20];

    const int t    = threadIdx.x;
    const int wave = t >> 5;
    const int lane = t & 31;
    const int half = lane >> 4;
    const int mrow = lane & 15;
    const int m0   = blockIdx.y * 16;
    const int n0w  = blockIdx.x * 64 + wave * 16;

    v8f acc = {};
    gemm_tile_acc<1024, 1024>(H2, Ws, Uu, m0, n0w, ldsA, ldsB[wave], acc);

    const float bv = bs[n0w + mrow];
#pragma unroll
    for (int r = 0; r < 8; ++r) {
        const int gm = m0 + r + 8 * half;
        Sout[(size_t)gm * Uu + n0w + mrow] = acc[r] + bv;
    }
}

// ---------------------------------------------------------------------------
// One-pass attention with online softmax (attention_vec read once: 256 MB,
// nontemporal so it doesn't thrash L2 holding the LSTM weights).
// One block (512 thr = 16 waves) per batch row; each wave streams T/16 steps
// keeping running (max, sum, rescaled ctx[32 f32/lane]); cross-wave combine
// via LDS + ds_add_f32. Emits both outputs:
//   out[0:BU] = h2,  out[BU:2BU] = c2 + ctx
// ---------------------------------------------------------------------------
__global__ void __launch_bounds__(512)
attention_kernel(const float* __restrict__ av, const float* __restrict__ S,
                 const float* __restrict__ H2, const float* __restrict__ C2,
                 float* __restrict__ out, int T, int Uu, int B)
{
    __shared__ float sS[1024];
    __shared__ float ctx_s[1024];
    __shared__ float wm[16], wl[16];

    const int b    = blockIdx.x;
    const int t    = threadIdx.x;
    const int wave = t >> 5;
    const int lane = t & 31;

    for (int u = t; u < Uu; u += 512) {
        sS[u]    = S[(size_t)b * Uu + u];
        ctx_s[u] = 0.0f;
    }
    __syncthreads();

    float m = -INFINITY, l = 0.0f;
    float ctx[32];
#pragma unroll
    for (int j = 0; j < 32; ++j) ctx[j] = 0.0f;

    const float* avb = av + (size_t)b * T * Uu;
    for (int tt = wave; tt < T; tt += 16) {
        const float* row = avb + (size_t)tt * Uu;
        __builtin_prefetch(avb + (size_t)(tt + 16) * Uu + lane * 32, 0, 0);

        v4f frag[8];
        float dot = 0.0f;
#pragma unroll
        for (int j = 0; j < 8; ++j) {
            const int u = (j * 32 + lane) * 4;
            frag[j] = __builtin_nontemporal_load((const v4f*)(row + u));
            dot += frag[j].x * sS[u] + frag[j].y * sS[u + 1] +
                   frag[j].z * sS[u + 2] + frag[j].w * sS[u + 3];
        }
#pragma unroll
        for (int off = 16; off > 0; off >>= 1) dot += __shfl_xor(dot, off, 32);

        const float mn    = fmaxf(m, dot);
        const float scale = __expf(m - mn);
        const float e     = __expf(dot - mn);
        l = l * scale + e;
#pragma unroll
        for (int j = 0; j < 8; ++j) {
            ctx[j * 4 + 0] = ctx[j * 4 + 0] * scale + e * frag[j].x;
            ctx[j * 4 + 1] = ctx[j * 4 + 1] * scale + e * frag[j].y;
            ctx[j * 4 + 2] = ctx[j * 4 + 2] * scale + e * frag[j].z;
            ctx[j * 4 + 3] = ctx[j * 4 + 3] * scale + e * frag[j].w;
        }
        m = mn;
    }

    if (lane == 0) { wm[wave] = m; wl[wave] = l; }
    __syncthreads();

    float M = wm[0];
#pragma unroll
    for (int w = 1; w < 16; ++w) M = fmaxf(M, wm[w]);
    float L = 0.0f;
#pragma unroll
    for (int w = 0; w < 16; ++w) L += wl[w] * __expf(wm[w] - M);

    const float f = __expf(m - M);
#pragma unroll
    for (int j = 0; j < 8; ++j) {
        const int u = (j * 32 + lane) * 4;
        atomicAdd(&ctx_s[u + 0], ctx[j * 4 + 0] * f);
        atomicAdd(&ctx_s[u + 1], ctx[j * 4 + 1] * f);
        atomicAdd(&ctx_s[u + 2], ctx[j * 4 + 2] * f);
        atomicAdd(&ctx_s[u + 3], ctx[j * 4 + 3] * f);
    }
    __syncthreads();

    const float invL = 1.0f / L;
    const size_t base = (size_t)b * Uu;
    for (int u = t; u < Uu; u += 512) {
        out[base + u]                  = H2[base + u];
        out[(size_t)B * Uu + base + u] = C2[base + u] + ctx_s[u] * invL;
    }
}

// ---------------------------------------------------------------------------
extern "C" void kernel_launch(void* const* d_in, const int* in_sizes, int n_in,
                              void* d_out, int out_size, void* d_ws, size_t ws_size,
                              hipStream_t stream)
{
    (void)in_sizes; (void)n_in; (void)out_size; (void)ws_size;
    // Reference shapes (fixed by harness)
    const int B = 128, T = 512, U = 1024;

    const float* x   = (const float*)d_in[0];
    const float* h0  = (const float*)d_in[1];
    const float* c0  = (const float*)d_in[2];
    const float* av  = (const float*)d_in[3];
    const float* W1  = (const float*)d_in[4];
    const float* U1  = (const float*)d_in[5];
    const float* b1  = (const float*)d_in[6];
    const float* W2  = (const float*)d_in[7];
    const float* U2  = (const float*)d_in[8];
    const float* b2  = (const float*)d_in[9];
    const float* Ws  = (const float*)d_in[10];
    const float* bs  = (const float*)d_in[11];
    float* out = (float*)d_out;

    float* ws = (float*)d_ws;
    const size_t BU = (size_t)B * U;
    float* h1 = ws;
    float* c1 = ws + BU;
    float* h2 = ws + 2 * BU;
    float* c2 = ws + 3 * BU;
    float* s  = ws + 4 * BU;

    dim3 gGemm(U / 16, B / 16);   // (64, 8)
    dim3 bGemm(128);

    // LSTM cell 1: z = x@W1 + h0@U1 + b1
    lstm_cell_kernel<<<gGemm, bGemm, 0, stream>>>(x, W1, h0, U1, b1, c0,
                                                  h1, c1, 1024, U, U);
    // LSTM cell 2 (faithful quirk): input=h1, h=h1, c=c1
    lstm_cell_kernel<<<gGemm, bGemm, 0, stream>>>(h1, W2, h1, U2, b2, c1,
                                                  h2, c2, U, U, U);
    // s = h2 @ Ws + bs
    sproj_kernel<<<dim3(U / 64, B / 16), bGemm, 0, stream>>>(h2, Ws, bs, s, U);
    // Fused attention + outputs
    attention_kernel<<<dim3(B), dim3(512), 0, stream>>>(av, s, h2, c2, out,
                                                        T, U, B);
}